// FusionCorrectionHead_17085379904362
// MI455X (gfx1250) — compile-verified
//
#include <hip/hip_runtime.h>
#include <math.h>

typedef float v2f __attribute__((ext_vector_type(2)));
typedef float v8f __attribute__((ext_vector_type(8)));

// ---------------------------------------------------------------------------
// Kernel 1: per-(b,j) heatmap max over 4096 floats -> sigmoid -> confidence.
// Memory-bound (88 MB total); coalesced float4 loads, wave32 shuffle reduce.
// ---------------------------------------------------------------------------
__global__ __launch_bounds__(256) void hm_conf_kernel(const float* __restrict__ hm,
                                                      float* __restrict__ conf) {
    const int bid = blockIdx.x;                 // 0 .. B*J-1
    const int t   = threadIdx.x;                // 0 .. 255
    const float4* p = (const float4*)(hm + (size_t)bid * 4096);

    float m = -3.402823466e38f;
#pragma unroll
    for (int i = 0; i < 4; ++i) {
        float4 v = p[t + i * 256];
        m = fmaxf(m, fmaxf(fmaxf(v.x, v.y), fmaxf(v.z, v.w)));
    }
    // wave32 reduction
#pragma unroll
    for (int off = 16; off > 0; off >>= 1)
        m = fmaxf(m, __shfl_xor(m, off, 32));

    __shared__ float smax[8];
    if ((t & 31) == 0) smax[t >> 5] = m;
    __syncthreads();
    if (t == 0) {
        float mm = smax[0];
#pragma unroll
        for (int i = 1; i < 8; ++i) mm = fmaxf(mm, smax[i]);
        conf[bid] = 1.0f / (1.0f + expf(-mm));   // exact sigmoid
    }
}

// exact GELU (approximate=False): 0.5*x*(1+erf(x/sqrt(2)))
__device__ __forceinline__ float gelu_exact(float x) {
    return 0.5f * x * (1.0f + erff(x * 0.70710678118654752440f));
}

// LayerNorm in-place over 64 features, 16 rows, 128 threads (8 per row).
__device__ __forceinline__ void ln_inplace(float h[16][65],
                                           const float* __restrict__ g,
                                           const float* __restrict__ be,
                                           float rsum[16][8], float rsq[16][8],
                                           int tid) {
    const int r = tid >> 3;       // row 0..15
    const int p = tid & 7;        // part 0..7
    float s = 0.f, ss = 0.f;
#pragma unroll
    for (int j = 0; j < 8; ++j) {
        float v = h[r][p * 8 + j];
        s += v; ss += v * v;
    }
    rsum[r][p] = s; rsq[r][p] = ss;
    __syncthreads();
    float ts = 0.f, tss = 0.f;
#pragma unroll
    for (int i = 0; i < 8; ++i) { ts += rsum[r][i]; tss += rsq[r][i]; }
    const float mu  = ts * (1.0f / 64.0f);
    const float var = tss * (1.0f / 64.0f) - mu * mu;
    const float sc  = rsqrtf(var + 1e-5f);
#pragma unroll
    for (int j = 0; j < 8; ++j) {
        const int c = p * 8 + j;
        h[r][c] = (h[r][c] - mu) * sc * g[c] + be[c];
    }
    __syncthreads();
}

// ---------------------------------------------------------------------------
// Kernel 2: fused MLP head. One 16-row tile per block, 4 waves (128 thr).
// fp32 WMMA (V_WMMA_F32_16X16X4_F32) -> exact fp32 arithmetic.
//   A-frag (16x4 f32): lane M = lane%16;  a[v] holds K = ks + 2*(lane/16) + v
//   B-frag (4x16 f32): lane N = lane%16;  b[v] holds K = ks + 2*(lane/16) + v
//   C/D   (16x16 f32): lane N = lane%16;  c[v] holds M = v + 8*(lane/16)
// ---------------------------------------------------------------------------
__global__ __launch_bounds__(128) void mlp_head_kernel(
    const float* __restrict__ fk, const float* __restrict__ dir,
    const float* __restrict__ conf,
    const float* __restrict__ W1, const float* __restrict__ b1,
    const float* __restrict__ g1, const float* __restrict__ be1,
    const float* __restrict__ W2, const float* __restrict__ b2,
    const float* __restrict__ g2, const float* __restrict__ be2,
    const float* __restrict__ W3, const float* __restrict__ b3,
    float* __restrict__ out, int nrows) {

    __shared__ float xs[16][12];     // input tile, K padded 10 -> 12
    __shared__ float w1s[12][64];    // W1 padded
    __shared__ float w2s[64][64];
    __shared__ float w3s[64][16];    // W3 padded 4 -> 16 cols
    __shared__ float h1[16][65];     // +1 pad: kills bank conflicts on A-frag reads
    __shared__ float h2[16][65];
    __shared__ float outs[16][16];
    __shared__ float rsum[16][8];
    __shared__ float rsq[16][8];

    const int tid  = threadIdx.x;
    const int lane = tid & 31;
    const int wave = tid >> 5;       // 0..3 -> owns N-cols [16*wave, 16*wave+16)
    const int half = lane >> 4;      // 0|1
    const int l16  = lane & 15;
    const int row0 = blockIdx.x * 16;

    // ---- stage weights & inputs into LDS -------------------------------
    for (int i = tid; i < 12 * 64; i += 128) {
        const int k = i >> 6, n = i & 63;
        w1s[k][n] = (k < 10) ? W1[k * 64 + n] : 0.0f;
    }
    for (int i = tid; i < 64 * 64; i += 128)
        w2s[i >> 6][i & 63] = W2[i];
    for (int i = tid; i < 64 * 16; i += 128) {
        const int k = i >> 4, n = i & 15;
        w3s[k][n] = (n < 4) ? W3[k * 4 + n] : 0.0f;
    }
    for (int i = tid; i < 16 * 12; i += 128) {
        const int r = i / 12, c = i % 12;
        const int grow = row0 + r;
        float v;
        if (c < 3)       v = fk[grow * 3 + c];
        else if (c < 6)  v = dir[grow * 3 + (c - 3)];
        else if (c < 9)  v = dir[grow * 3 + (c - 6)] - fk[grow * 3 + (c - 6)];
        else if (c == 9) v = conf[grow];
        else             v = 0.0f;
        xs[r][c] = v;
    }
    __syncthreads();

    // ---- Layer 1: (16x12) @ (12x64), K=12 -> 3 WMMAs per wave ----------
    {
        v8f c = {0.f, 0.f, 0.f, 0.f, 0.f, 0.f, 0.f, 0.f};
        const int ncol = wave * 16 + l16;
#pragma unroll
        for (int ks = 0; ks < 12; ks += 4) {
            v2f a, b;
            a[0] = xs[l16][ks + 2 * half + 0];
            a[1] = xs[l16][ks + 2 * half + 1];
            b[0] = w1s[ks + 2 * half + 0][ncol];
            b[1] = w1s[ks + 2 * half + 1][ncol];
            c = __builtin_amdgcn_wmma_f32_16x16x4_f32(false, a, false, b,
                                                      (short)0, c, false, false);
        }
        const float bias = b1[ncol];
#pragma unroll
        for (int v = 0; v < 8; ++v)
            h1[v + 8 * half][ncol] = gelu_exact(c[v] + bias);
    }
    __syncthreads();
    ln_inplace(h1, g1, be1, rsum, rsq, tid);

    // ---- Layer 2: (16x64) @ (64x64), K=64 -> 16 WMMAs per wave ---------
    {
        v8f c = {0.f, 0.f, 0.f, 0.f, 0.f, 0.f, 0.f, 0.f};
        const int ncol = wave * 16 + l16;
#pragma unroll
        for (int ks = 0; ks < 64; ks += 4) {
            v2f a, b;
            a[0] = h1[l16][ks + 2 * half + 0];
            a[1] = h1[l16][ks + 2 * half + 1];
            b[0] = w2s[ks + 2 * half + 0][ncol];
            b[1] = w2s[ks + 2 * half + 1][ncol];
            c = __builtin_amdgcn_wmma_f32_16x16x4_f32(false, a, false, b,
                                                      (short)0, c, false, false);
        }
        const float bias = b2[ncol];
#pragma unroll
        for (int v = 0; v < 8; ++v)
            h2[v + 8 * half][ncol] = gelu_exact(c[v] + bias);
    }
    __syncthreads();
    ln_inplace(h2, g2, be2, rsum, rsq, tid);

    // ---- Layer 3: (16x64) @ (64x16 zero-padded), wave 0 only -----------
    if (wave == 0) {
        v8f c = {0.f, 0.f, 0.f, 0.f, 0.f, 0.f, 0.f, 0.f};
#pragma unroll
        for (int ks = 0; ks < 64; ks += 4) {
            v2f a, b;
            a[0] = h2[l16][ks + 2 * half + 0];
            a[1] = h2[l16][ks + 2 * half + 1];
            b[0] = w3s[ks + 2 * half + 0][l16];
            b[1] = w3s[ks + 2 * half + 1][l16];
            c = __builtin_amdgcn_wmma_f32_16x16x4_f32(false, a, false, b,
                                                      (short)0, c, false, false);
        }
#pragma unroll
        for (int v = 0; v < 8; ++v)
            outs[v + 8 * half][l16] = c[v];
    }
    __syncthreads();

    // ---- Epilogue: one thread per row ----------------------------------
    if (tid < 16) {
        const int grow = row0 + tid;
        const float d0 = outs[tid][0] + b3[0];
        const float d1 = outs[tid][1] + b3[1];
        const float d2 = outs[tid][2] + b3[2];
        const float o3 = outs[tid][3] + b3[3];
        const float alpha = 1.0f / (1.0f + expf(-o3));

        const float fx = fk[grow * 3 + 0], fy = fk[grow * 3 + 1], fz = fk[grow * 3 + 2];
        const float dx = dir[grow * 3 + 0], dy = dir[grow * 3 + 1], dz = dir[grow * 3 + 2];

        // output layout: kp3d_final[nrows*3] | alpha[nrows] | delta[nrows*3]
        float* kp_final = out;
        float* p_alpha  = out + (size_t)nrows * 3;
        float* p_delta  = out + (size_t)nrows * 4;

        kp_final[grow * 3 + 0] = alpha * dx + (1.0f - alpha) * fx + d0;
        kp_final[grow * 3 + 1] = alpha * dy + (1.0f - alpha) * fy + d1;
        kp_final[grow * 3 + 2] = alpha * dz + (1.0f - alpha) * fz + d2;
        p_alpha[grow] = alpha;
        p_delta[grow * 3 + 0] = d0;
        p_delta[grow * 3 + 1] = d1;
        p_delta[grow * 3 + 2] = d2;
    }
}

extern "C" void kernel_launch(void* const* d_in, const int* in_sizes, int n_in,
                              void* d_out, int out_size, void* d_ws, size_t ws_size,
                              hipStream_t stream) {
    const float* fk  = (const float*)d_in[0];
    const float* dir = (const float*)d_in[1];
    const float* hm  = (const float*)d_in[2];
    const float* W1  = (const float*)d_in[3];
    const float* b1  = (const float*)d_in[4];
    const float* g1  = (const float*)d_in[5];
    const float* be1 = (const float*)d_in[6];
    const float* W2  = (const float*)d_in[7];
    const float* b2  = (const float*)d_in[8];
    const float* g2  = (const float*)d_in[9];
    const float* be2 = (const float*)d_in[10];
    const float* W3  = (const float*)d_in[11];
    const float* b3  = (const float*)d_in[12];
    float* out  = (float*)d_out;
    float* conf = (float*)d_ws;                 // B*J floats of scratch

    const int nrows = in_sizes[0] / 3;          // B*J = 5376

    hm_conf_kernel<<<nrows, 256, 0, stream>>>(hm, conf);
    mlp_head_kernel<<<nrows / 16, 128, 0, stream>>>(fk, dir, conf,
                                                    W1, b1, g1, be1,
                                                    W2, b2, g2, be2,
                                                    W3, b3, out, nrows);
}